// MultiScaleFSQTP_88708254532122
// MI455X (gfx1250) — compile-verified
//
#include <hip/hip_runtime.h>
#include <hip/hip_bf16.h>
#include <math.h>

// ---------------------------------------------------------------------------
// MultiScale FSQ (B=4, C=64, T=5, H=W=64), f32 throughout.
// All down/upsample contractions run on V_WMMA_F32_16X16X4_F32 (wave32).
// Inner loops are branchless: weights zero-padded to K4=ceil4(K), row/col
// indices clamped once, stores guarded only in the epilogue.
// ---------------------------------------------------------------------------

typedef __attribute__((ext_vector_type(2))) float v2f;
typedef __attribute__((ext_vector_type(8))) float v8f;

#define BB 4
#define CC 64
#define TT 5
#define HH 64
#define WW 64
#define N_FULL (BB * CC * TT * HH * WW) // 5,242,880 floats

// workspace layout (in floats); weight rows are stored with stride K4 (zero-padded)
#define WT_OFF 0        // area   Wt: t rows x stride 8    (<=64)
#define WH_OFF 64       // area   Wh: h rows x stride 64   (<=4096)
#define WW_OFF 4160     // area   Ww: w rows x stride 64   (<=4096)
#define UT_OFF 8256     // interp Ut: 5 rows x stride t4   (<=64)
#define UH_OFF 8320     // interp Uh: 64 rows x stride h4  (<=4096)
#define UW_OFF 12416    // interp Uw: 64 rows x stride w4  (<=4096)
#define WGT_FLOATS 16512
#define R_OFF WGT_FLOATS
#define A_OFF (R_OFF + N_FULL)
#define A_FLOATS 4194304
#define B_OFF (A_OFF + A_FLOATS)

static __host__ __device__ inline int ceil4(int x) { return (x + 3) & ~3; }

// ---------------------------------------------------------------------------
// init: residual = x, out = 0
__global__ void msfsq_init_kernel(const float* __restrict__ x,
                                  float* __restrict__ resid,
                                  float* __restrict__ out, int n) {
  int i = blockIdx.x * blockDim.x + threadIdx.x;
  if (i < n) { resid[i] = x[i]; out[i] = 0.0f; }
}

// final identity stage: out += fsq(residual)
__global__ void msfsq_final_kernel(const float* __restrict__ resid,
                                   float* __restrict__ out, int n) {
  int i = blockIdx.x * blockDim.x + threadIdx.x;
  if (i < n) {
    float b = tanhf(resid[i]);
    float q = rintf(b * 4.0f) * 0.25f;
    out[i] += q;
  }
}

// ---------------------------------------------------------------------------
// build per-stage weight matrices on device (area-pool + half-pixel linear),
// each row zero-padded to its K4 stride so GEMM K-loops need no bounds tests.
__device__ __forceinline__ float area_w(int i, int k, int n, int N) {
  int s = (i * N) / n;
  int e = ((i + 1) * N + n - 1) / n;
  return (k >= s && k < e) ? (1.0f / (float)(e - s)) : 0.0f;
}
__device__ __forceinline__ float lin_w(int I, int k, int Nout, int nin) {
  float pos = (I + 0.5f) * (float)nin / (float)Nout - 0.5f;
  float fl = floorf(pos);
  int i0 = (int)fl;
  float frac = pos - fl;
  int ia = i0 < 0 ? 0 : (i0 > nin - 1 ? nin - 1 : i0);
  int i1 = i0 + 1;
  int ib = i1 < 0 ? 0 : (i1 > nin - 1 ? nin - 1 : i1);
  float v = 0.0f;
  if (k == ia) v += 1.0f - frac;
  if (k == ib) v += frac;
  return v;
}

__global__ void msfsq_weights_kernel(float* __restrict__ wb, int t, int h, int w) {
  int tid = threadIdx.x, nt = blockDim.x;
  int t4 = ceil4(t), h4 = ceil4(h), w4 = ceil4(w);
  // area matrices: (rows, K4) with K real = TT/HH/WW
  for (int i = tid; i < t * 8; i += nt) {
    int r = i / 8, k = i % 8;
    wb[WT_OFF + i] = (k < TT) ? area_w(r, k, t, TT) : 0.0f;
  }
  for (int i = tid; i < h * HH; i += nt) wb[WH_OFF + i] = area_w(i / HH, i % HH, h, HH);
  for (int i = tid; i < w * WW; i += nt) wb[WW_OFF + i] = area_w(i / WW, i % WW, w, WW);
  // interp matrices: (rows, K4) with K real = t/h/w
  for (int i = tid; i < TT * t4; i += nt) {
    int r = i / t4, k = i % t4;
    wb[UT_OFF + i] = (k < t) ? lin_w(r, k, TT, t) : 0.0f;
  }
  for (int i = tid; i < HH * h4; i += nt) {
    int r = i / h4, k = i % h4;
    wb[UH_OFF + i] = (k < h) ? lin_w(r, k, HH, h) : 0.0f;
  }
  for (int i = tid; i < WW * w4; i += nt) {
    int r = i / w4, k = i % w4;
    wb[UW_OFF + i] = (k < w) ? lin_w(r, k, WW, w) : 0.0f;
  }
}

// ---------------------------------------------------------------------------
// Pattern A: contract innermost dim.  out[m,j] = sum_k in[m*K+k] * wm[j*K4+k]
// wm rows are zero-padded to K4, so the loop runs k0=0..K4 branch-free.
// in over-reads <=3 floats past its last row (multiplied by zero weights);
// all pattern-A inputs live in oversized workspace slabs, so this is safe.
__global__ void msfsq_gemm_inner_kernel(const float* __restrict__ in,
                                        const float* __restrict__ wm,
                                        float* __restrict__ out,
                                        int M, int K, int K4, int N) {
  int wave = blockIdx.x * (blockDim.x >> 5) + (threadIdx.x >> 5);
  int lane = threadIdx.x & 31;
  int mtiles = (M + 15) >> 4;
  int ntiles = (N + 15) >> 4;
  if (wave >= mtiles * ntiles) return;
  int m0 = (wave % mtiles) << 4;
  int j0 = (wave / mtiles) << 4;
  int lrow = lane & 15;
  int hi = lane >> 4; // 0 or 1
  int m = m0 + lrow;
  int j = j0 + lrow;
  int mL = (m < M) ? m : (M - 1);
  int jL = (j < N) ? j : (N - 1);
  const float* __restrict__ ain = in + (size_t)mL * K + 2 * hi;
  const float* __restrict__ bwm = wm + (size_t)jL * K4 + 2 * hi; // 8B aligned
  v8f acc = {};
  for (int k0 = 0; k0 < K4; k0 += 4) {
    v2f a, b;
    a.x = ain[k0];
    a.y = ain[k0 + 1];
    v2f wv = *(const v2f*)(bwm + k0);
    b.x = wv.x;
    b.y = wv.y;
    acc = __builtin_amdgcn_wmma_f32_16x16x4_f32(false, a, false, b,
                                                (short)0, acc, false, false);
  }
  if (j < N) {
#pragma unroll
    for (int r = 0; r < 8; ++r) {
      int mr = m0 + r + 8 * hi;
      if (mr < M) out[(size_t)mr * N + j] = acc[r];
    }
  }
}

// ---------------------------------------------------------------------------
// Pattern B: contract middle dim. out[mo,j,p] = sum_k wm[j*K4+k] * in[(mo*K+k)*P+p]
// Weight operand zero-padded to K4; input k clamped to K-1 (values for k>=K
// are multiplied by zero weights). All loads unconditional.
// quant: FSQ epilogue (last downsample op). fuse: out_acc+=v; resid-=v.
__global__ void msfsq_gemm_mid_kernel(const float* __restrict__ in,
                                      const float* __restrict__ wm,
                                      float* __restrict__ out,
                                      float* __restrict__ out_acc,
                                      float* __restrict__ resid,
                                      int Mo, int K, int K4, int N, int P,
                                      int quant, int fuse) {
  int wave = blockIdx.x * (blockDim.x >> 5) + (threadIdx.x >> 5);
  int lane = threadIdx.x & 31;
  int ntiles = (N + 15) >> 4;
  int ptiles = (P + 15) >> 4;
  if (wave >= Mo * ntiles * ptiles) return;
  int tp = wave % ptiles;
  int rest = wave / ptiles;
  int j0 = (rest % ntiles) << 4;
  int mo = rest / ntiles;
  int p0 = tp << 4;
  int lrow = lane & 15;
  int hi = lane >> 4;
  int j = j0 + lrow;
  int p = p0 + lrow;
  int jL = (j < N) ? j : (N - 1);
  int pL = (p < P) ? p : (P - 1);
  const float* __restrict__ awm = wm + (size_t)jL * K4 + 2 * hi; // 8B aligned
  const float* __restrict__ inb = in + (size_t)mo * K * P + pL;
  v8f acc = {};
  for (int k0 = 0; k0 < K4; k0 += 4) {
    int ka = k0 + 2 * hi;
    int kb = ka + 1;
    int kac = (ka < K) ? ka : (K - 1);
    int kbc = (kb < K) ? kb : (K - 1);
    v2f a, b;
    v2f wv = *(const v2f*)(awm + k0);
    a.x = wv.x;
    a.y = wv.y;
    b.x = inb[(size_t)kac * P];
    b.y = inb[(size_t)kbc * P];
    acc = __builtin_amdgcn_wmma_f32_16x16x4_f32(false, a, false, b,
                                                (short)0, acc, false, false);
  }
  if (p < P) {
#pragma unroll
    for (int r = 0; r < 8; ++r) {
      int jr = j0 + r + 8 * hi;
      if (jr < N) {
        float v = acc[r];
        if (quant) { v = tanhf(v); v = rintf(v * 4.0f) * 0.25f; }
        size_t idx = ((size_t)mo * N + jr) * P + p;
        if (fuse) { out_acc[idx] += v; resid[idx] -= v; }
        else      { out[idx] = v; }
      }
    }
  }
}

// ---------------------------------------------------------------------------
static inline int cdiv_i(int a, int b) { return (a + b - 1) / b; }

extern "C" void kernel_launch(void* const* d_in, const int* in_sizes, int n_in,
                              void* d_out, int out_size, void* d_ws, size_t ws_size,
                              hipStream_t stream) {
  (void)in_sizes; (void)n_in; (void)out_size; (void)ws_size;
  const float* x = (const float*)d_in[0];
  float* out = (float*)d_out;
  float* wsf = (float*)d_ws;
  float* wgt = wsf;           // weight matrices
  float* R   = wsf + R_OFF;   // residual (full size)
  float* Abuf = wsf + A_OFF;  // scratch A
  float* Bbuf = wsf + B_OFF;  // scratch B

  const int THREADS = 256, WPB = THREADS / 32;

  // residual = x, out = 0
  msfsq_init_kernel<<<cdiv_i(N_FULL, THREADS), THREADS, 0, stream>>>(x, R, out, N_FULL);

  static const int HS[16] = {1,2,3,4,5,6,8,10,13,16,20,24,32,40,48,64};
  static const int TS[16] = {1,2,3,5,5,5,5,5,5,5,5,5,5,5,5,5};

  for (int s = 0; s < 15; ++s) { // stage 15 (identity) handled after loop
    const int t = TS[s], h = HS[s], w = HS[s];
    const int t4 = ceil4(t), h4 = ceil4(h), w4 = ceil4(w);

    // per-stage weight matrices (zero-padded rows)
    msfsq_weights_kernel<<<1, THREADS, 0, stream>>>(wgt, t, h, w);

    // ---- downsample -------------------------------------------------------
    // W: R (M=B*C*T*H, K=64) x Ww(w,64) -> Abuf (B,C,T,H,w)
    {
      int M = BB * CC * TT * HH, K = WW, N = w;
      int tiles = cdiv_i(M, 16) * cdiv_i(N, 16);
      msfsq_gemm_inner_kernel<<<cdiv_i(tiles, WPB), THREADS, 0, stream>>>(
          R, wgt + WW_OFF, Abuf, M, K, K, N);
    }
    // H: Abuf (Mo=B*C*T, K=64, P=w) x Wh(h,64) -> Bbuf (B,C,T,h,w)
    {
      int Mo = BB * CC * TT, K = HH, N = h, P = w;
      int quant = (t == TT) ? 1 : 0; // last downsample op when no T reduction
      int tiles = Mo * cdiv_i(N, 16) * cdiv_i(P, 16);
      msfsq_gemm_mid_kernel<<<cdiv_i(tiles, WPB), THREADS, 0, stream>>>(
          Abuf, wgt + WH_OFF, Bbuf, nullptr, nullptr, Mo, K, K, N, P, quant, 0);
    }
    float* z;
    if (t < TT) {
      // T: Bbuf (Mo=B*C, K=5, P=h*w) x Wt(t,5 pad 8) -> Abuf (B,C,t,h,w), quantized
      int Mo = BB * CC, K = TT, N = t, P = h * w;
      int tiles = Mo * cdiv_i(N, 16) * cdiv_i(P, 16);
      msfsq_gemm_mid_kernel<<<cdiv_i(tiles, WPB), THREADS, 0, stream>>>(
          Bbuf, wgt + WT_OFF, Abuf, nullptr, nullptr, Mo, K, 8, N, P, 1, 0);
      z = Abuf;
    } else {
      z = Bbuf;
    }

    // ---- upsample (trilinear, half-pixel) --------------------------------
    float* upw = (z == Abuf) ? Bbuf : Abuf; // the other scratch buffer
    // W: z (M=B*C*t*h, K=w) x Uw(64, w pad w4) -> upw (B,C,t,h,64)
    {
      int M = BB * CC * t * h, K = w, N = WW;
      int tiles = cdiv_i(M, 16) * cdiv_i(N, 16);
      msfsq_gemm_inner_kernel<<<cdiv_i(tiles, WPB), THREADS, 0, stream>>>(
          z, wgt + UW_OFF, upw, M, K, w4, N);
    }
    if (t == TT) {
      // H: upw (Mo=B*C*T, K=h, P=64) x Uh(64, h pad h4) -> fused into (out, R)
      int Mo = BB * CC * TT, K = h, N = HH, P = WW;
      int tiles = Mo * cdiv_i(N, 16) * cdiv_i(P, 16);
      msfsq_gemm_mid_kernel<<<cdiv_i(tiles, WPB), THREADS, 0, stream>>>(
          upw, wgt + UH_OFF, nullptr, out, R, Mo, K, h4, N, P, 0, 1);
    } else {
      float* uph = z; // z buffer is free now
      // H: upw (Mo=B*C*t, K=h, P=64) -> uph (B,C,t,64,64)
      {
        int Mo = BB * CC * t, K = h, N = HH, P = WW;
        int tiles = Mo * cdiv_i(N, 16) * cdiv_i(P, 16);
        msfsq_gemm_mid_kernel<<<cdiv_i(tiles, WPB), THREADS, 0, stream>>>(
            upw, wgt + UH_OFF, uph, nullptr, nullptr, Mo, K, h4, N, P, 0, 0);
      }
      // T: uph (Mo=B*C, K=t, P=64*64) x Ut(5, t pad t4) -> fused into (out, R)
      {
        int Mo = BB * CC, K = t, N = TT, P = HH * WW;
        int tiles = Mo * cdiv_i(N, 16) * cdiv_i(P, 16);
        msfsq_gemm_mid_kernel<<<cdiv_i(tiles, WPB), THREADS, 0, stream>>>(
            uph, wgt + UT_OFF, nullptr, out, R, Mo, K, t4, N, P, 0, 1);
      }
    }
  }

  // stage 15: identity resolution -> out += fsq(residual)
  msfsq_final_kernel<<<cdiv_i(N_FULL, THREADS), THREADS, 0, stream>>>(R, out, N_FULL);
}